// symLoss_18236431139042
// MI455X (gfx1250) — compile-verified
//
#include <hip/hip_runtime.h>
#include <stdint.h>
#include <math.h>

#define GRID_SZ   32
#define G3        (GRID_SZ * GRID_SZ * GRID_SZ)   // 32768
#define TPB       256
#define PTS_BLK   256

// 3-float aggregate with 4-byte alignment so the backend can emit a single
// b96 load at (idx*12)-byte offsets (dword-aligned only).
struct __attribute__((aligned(4))) f3 { float x, y, z; };

// round(clip(x,0,31)) of (x - grid_min)*32, grid_min = -0.5 + 0.5/32
__device__ __forceinline__ int cell_index(float x) {
    const float gm = -0.484375f;               // exactly representable
    float v = (x - gm) * 32.0f;
    v = fminf(fmaxf(v, 0.0f), 31.0f);
    return (int)rintf(v);                      // v_rndne_f32: round half to even == jnp.round
}

__global__ __launch_bounds__(TPB) void sym_loss_partial(
    const float* __restrict__ points,   // (B, N, 3)
    const float* __restrict__ cp,       // (B, G3, 3)
    const float* __restrict__ voxel,    // (B, G3)
    const float* __restrict__ plane,    // (3, 4)
    const float* __restrict__ quat,     // (3, 4)
    float* __restrict__ partial,        // (B*blks_per_batch, 2)
    int n_points)
{
    __shared__ float s_pts[PTS_BLK * 3];       // 3 KB staged chunk of points
    __shared__ float s_red[16];

    const int tid   = threadIdx.x;
    const int chunk = blockIdx.x;              // 2D grid: no integer division
    const int b     = blockIdx.y;
    const int blk   = b * gridDim.x + chunk;

    // ---- async global -> LDS stage of this block's 256 points (3072 bytes) ----
    // Each thread async-copies 3 dwords; INST_OFFSET applies to both the LDS and
    // global addresses, so one vdst/vaddr pair covers the whole chunk.
    const float* gsrc = points + ((size_t)b * n_points + (size_t)chunk * PTS_BLK) * 3;
    uint32_t ldsa = (uint32_t)(size_t)(&s_pts[0]) + (uint32_t)tid * 4u;  // low 32b of flat = LDS offset
    uint32_t voff = (uint32_t)tid * 4u;
    asm volatile(
        "global_load_async_to_lds_b32 %0, %1, %2 offset:0\n\t"
        "global_load_async_to_lds_b32 %0, %1, %2 offset:1024\n\t"
        "global_load_async_to_lds_b32 %0, %1, %2 offset:2048\n\t"
        :: "v"(ldsa), "v"(voff), "s"(gsrc) : "memory");
    asm volatile("s_wait_asynccnt 0" ::: "memory");
    __syncthreads();

    const f3 p  = *(const f3*)&s_pts[tid * 3];  // single ds_load_b96
    const float px = p.x, py = p.y, pz = p.z;

    float ox[6], oy[6], oz[6];

    // ---- 3 plane reflections: p - 2*((p.n + d)/(n.n))*n ----
    #pragma unroll
    for (int i = 0; i < 3; ++i) {
        float n0 = plane[i * 4 + 0], n1 = plane[i * 4 + 1];
        float n2 = plane[i * 4 + 2], d  = plane[i * 4 + 3];
        float s    = n0 * n0 + n1 * n1 + n2 * n2;
        float dist = (px * n0 + py * n1 + pz * n2 + d) / s;
        float k    = 2.0f * dist;
        ox[i] = px - k * n0;
        oy[i] = py - k * n1;
        oz[i] = pz - k * n2;
    }

    // ---- 3 quaternion rotations: p + w*t + cross(u,t), t = 2*cross(u,p) ----
    #pragma unroll
    for (int i = 0; i < 3; ++i) {
        float q0 = quat[i * 4 + 0], q1 = quat[i * 4 + 1];
        float q2 = quat[i * 4 + 2], q3 = quat[i * 4 + 3];
        float inv = 1.0f / sqrtf(q0 * q0 + q1 * q1 + q2 * q2 + q3 * q3);
        float w  = q0 * inv;
        float u0 = q1 * inv, u1 = q2 * inv, u2 = q3 * inv;
        float tx = 2.0f * (u1 * pz - u2 * py);
        float ty = 2.0f * (u2 * px - u0 * pz);
        float tz = 2.0f * (u0 * py - u1 * px);
        float cx = u1 * tz - u2 * ty;
        float cy = u2 * tx - u0 * tz;
        float cz = u0 * ty - u1 * tx;
        ox[3 + i] = px + w * tx + cx;
        oy[3 + i] = py + w * ty + cy;
        oz[3 + i] = pz + w * tz + cz;
    }

    // ---- all 6 cell indices up front so the 12 gathers can overlap ----
    int idx[6];
    #pragma unroll
    for (int i = 0; i < 6; ++i)
        idx[i] = cell_index(ox[i]) * 1024 + cell_index(oy[i]) * 32 + cell_index(oz[i]);

    const f3*    cpb = (const f3*)(cp + (size_t)b * G3 * 3);
    const float* vxb = voxel + (size_t)b * G3;

    float aref = 0.0f, arot = 0.0f;
    #pragma unroll
    for (int i = 0; i < 6; ++i) {
        int id   = idx[i];
        float m  = 1.0f - vxb[id];       // global_load_b32 gather (L2-resident)
        f3 c     = cpb[id];              // single global_load_b96 gather
        float dx = (ox[i] - c.x) * m;
        float dy = (oy[i] - c.y) * m;
        float dz = (oz[i] - c.z) * m;
        float v  = dx * dx + dy * dy + dz * dz;
        if (i < 3) aref += v; else arot += v;
    }

    // ---- wave32 shuffle reduction, then cross-wave via LDS ----
    #pragma unroll
    for (int off = 16; off > 0; off >>= 1) {
        aref += __shfl_down(aref, off, 32);
        arot += __shfl_down(arot, off, 32);
    }
    const int wave = tid >> 5, lane = tid & 31;
    if (lane == 0) { s_red[wave] = aref; s_red[8 + wave] = arot; }
    __syncthreads();
    if (tid == 0) {
        float r = 0.0f, t = 0.0f;
        #pragma unroll
        for (int w = 0; w < 8; ++w) { r += s_red[w]; t += s_red[8 + w]; }
        partial[(size_t)blk * 2 + 0] = r;
        partial[(size_t)blk * 2 + 1] = t;
    }
}

__global__ __launch_bounds__(TPB) void sym_loss_reduce(
    const float* __restrict__ partial, float* __restrict__ out,
    int nblk, float inv_batch)
{
    __shared__ float s_red[16];
    float r = 0.0f, t = 0.0f;
    for (int i = threadIdx.x; i < nblk; i += TPB) {
        r += partial[(size_t)i * 2 + 0];
        t += partial[(size_t)i * 2 + 1];
    }
    #pragma unroll
    for (int off = 16; off > 0; off >>= 1) {
        r += __shfl_down(r, off, 32);
        t += __shfl_down(t, off, 32);
    }
    const int wave = threadIdx.x >> 5, lane = threadIdx.x & 31;
    if (lane == 0) { s_red[wave] = r; s_red[8 + wave] = t; }
    __syncthreads();
    if (threadIdx.x == 0) {
        float rr = 0.0f, tt = 0.0f;
        #pragma unroll
        for (int w = 0; w < 8; ++w) { rr += s_red[w]; tt += s_red[8 + w]; }
        out[0] = rr * inv_batch;   // ref_loss
        out[1] = tt * inv_batch;   // rot_loss
    }
}

extern "C" void kernel_launch(void* const* d_in, const int* in_sizes, int n_in,
                              void* d_out, int out_size, void* d_ws, size_t ws_size,
                              hipStream_t stream) {
    const float* points = (const float*)d_in[0];
    const float* cp     = (const float*)d_in[1];
    const float* voxel  = (const float*)d_in[2];
    const float* plane  = (const float*)d_in[3];
    const float* quat   = (const float*)d_in[4];
    float*       out    = (float*)d_out;
    float*       ws     = (float*)d_ws;

    const int B = in_sizes[2] / G3;                 // voxel is (B, 32768)
    const int N = in_sizes[0] / (B * 3);            // points is (B, N, 3)
    const int blks_per_batch = N / PTS_BLK;         // N=4096 -> 16
    const int nblk = B * blks_per_batch;            // 4096

    dim3 grid(blks_per_batch, B, 1);
    sym_loss_partial<<<grid, TPB, 0, stream>>>(points, cp, voxel, plane, quat, ws, N);
    sym_loss_reduce<<<1, TPB, 0, stream>>>(ws, out, nblk, 1.0f / (float)B);
}